// EncoderGCN3_75265006895440
// MI455X (gfx1250) — compile-verified
//
#include <hip/hip_runtime.h>
#include <math.h>

typedef __attribute__((ext_vector_type(2))) float v2f;
typedef __attribute__((ext_vector_type(8))) float v8f;

#define NN 10000
#define WPB 8      // waves per block (256 threads, wave32)
#define KDIM 256   // inner dimension is 256 for all three layers

// ---------------- degree / normalization ----------------

__global__ void deg_init_kernel(float* deg, int n) {
    int i = blockIdx.x * blockDim.x + threadIdx.x;
    if (i < n) deg[i] = 1.0f;  // self-loop
}

__global__ void deg_edge_kernel(const int* __restrict__ dst, float* deg, int E) {
    int e = blockIdx.x * blockDim.x + threadIdx.x;
    if (e < E) atomicAdd(&deg[dst[e]], 1.0f);
}

__global__ void dinv_kernel(float* deg, int n) {
    int i = blockIdx.x * blockDim.x + threadIdx.x;
    if (i < n) {
        float d = deg[i];
        deg[i] = d > 0.0f ? rsqrtf(d) : 0.0f;  // in-place deg -> d^-1/2
    }
}

// ---------------- dense transform: H = (relu?)A @ W via V_WMMA_F32_16X16X4_F32 ----------------
// Block = 8 waves sharing one 256x16 B tile staged in LDS (16 KB); each wave
// computes one 16x16 tile of H (8 consecutive row tiles per block).
// LDS layout [k][c]: lane (r, half) reads dword ko*16+r -> bank r+32*half, conflict-free.

template <int RELU>
__global__ void gemm_wmma_f32_kernel(const float* __restrict__ A,   // M x 256
                                     const float* __restrict__ W,   // 256 x N
                                     float* __restrict__ H,         // M x N
                                     int M, int N) {
    __shared__ float ldsB[KDIM * 16];  // 16 KB

    const int tiles_n = N >> 4;
    const int tmb = blockIdx.x / tiles_n;       // group of WPB row tiles
    const int tn  = blockIdx.x - tmb * tiles_n; // shared column tile

    // cooperative B-tile load: 256x16 floats as float4 (4 per thread)
    for (int i = threadIdx.x; i < (KDIM * 16) / 4; i += blockDim.x) {
        const int k  = i >> 2;
        const int c4 = i & 3;
        float4 v = *(const float4*)(W + (size_t)k * N + tn * 16 + c4 * 4);
        *(float4*)(ldsB + k * 16 + c4 * 4) = v;
    }
    __syncthreads();

    const int wave = threadIdx.x >> 5;
    const int lane = threadIdx.x & 31;
    const int half = lane >> 4;   // K-pair {0,1} vs {2,3} within a 4-wide K step
    const int r    = lane & 15;   // row (A) / col (B) within tile

    const int tm = tmb * WPB + wave;
    if (tm * 16 >= M) return;     // uniform per wave: EXEC all-ones for WMMA

    const float* Arow = A + (size_t)(tm * 16 + r) * KDIM;

    v8f c = {};
    #pragma unroll 8
    for (int k = 0; k < KDIM; k += 4) {
        const int ko = k + 2 * half;           // even -> float2-aligned
        v2f a = *(const v2f*)(Arow + ko);
        if (RELU) {
            a.x = fmaxf(a.x, 0.0f);
            a.y = fmaxf(a.y, 0.0f);
        }
        v2f b;
        b.x = ldsB[ko * 16 + r];
        b.y = ldsB[(ko + 1) * 16 + r];
        // 8 args: (neg_a, A, neg_b, B, c_mod, C, reuse_a, reuse_b)
        c = __builtin_amdgcn_wmma_f32_16x16x4_f32(false, a, false, b,
                                                  (short)0, c, false, false);
    }

    // D layout: VGPR i -> row (i + 8*half), col r
    float* Hp = H + (size_t)(tm * 16 + 8 * half) * N + tn * 16 + r;
    #pragma unroll
    for (int i = 0; i < 8; ++i) Hp[(size_t)i * N] = c[i];
}

// ---------------- out = h * dinv^2 + bias (self-loop term; fully overwrites out) ----------------

template <int F>
__global__ void self_bias_kernel(const float* __restrict__ h, const float* __restrict__ dinv,
                                 const float* __restrict__ bias, float* __restrict__ out,
                                 int n) {
    int idx = blockIdx.x * blockDim.x + threadIdx.x;
    if (idx < n * F) {
        int i = idx / F;          // F power of two -> shift
        int f = idx & (F - 1);
        float di = dinv[i];
        out[idx] = h[idx] * di * di + bias[f];
    }
}

// ---------------- edge scatter: out[dst] += h[src] * dinv[src]*dinv[dst] ----------------
// F/4 lanes per edge, float4 gathers from L2-resident h, f32 atomics into L2-resident out.

template <int F>
__global__ void scatter_kernel(const int* __restrict__ src, const int* __restrict__ dst,
                               const float* __restrict__ dinv, const float* __restrict__ h,
                               float* __restrict__ out, int E) {
    constexpr int TPE = F >> 2;                   // threads per edge (float4 lanes)
    constexpr int EPB = 256 / TPE;                // edges per block
    const int e  = blockIdx.x * EPB + (int)(threadIdx.x / TPE);
    const int f4 = threadIdx.x & (TPE - 1);
    if (e >= E) return;
    const int s = src[e];
    const int d = dst[e];
    const float nrm = dinv[s] * dinv[d];
    const float4 hv = *(const float4*)(h + (size_t)s * F + f4 * 4);
    float* o = out + (size_t)d * F + f4 * 4;
    atomicAdd(o + 0, hv.x * nrm);
    atomicAdd(o + 1, hv.y * nrm);
    atomicAdd(o + 2, hv.z * nrm);
    atomicAdd(o + 3, hv.w * nrm);
}

// ---------------- orchestration ----------------

static inline int gemm_blocks(int M, int N) {
    return ((M / 16 + WPB - 1) / WPB) * (N / 16);
}

static void gcn_graph(const float* X, const int* ei, int E,
                      const float* W1, const float* b1,
                      const float* W2, const float* b2,
                      const float* W3, const float* b3,
                      float* out, float* buf0, float* buf1, float* dinv,
                      hipStream_t stream) {
    const int* esrc = ei;
    const int* edst = ei + E;

    // degrees -> dinv (once per graph)
    deg_init_kernel<<<(NN + 255) / 256, 256, 0, stream>>>(dinv, NN);
    deg_edge_kernel<<<(E + 255) / 256, 256, 0, stream>>>(edst, dinv, E);
    dinv_kernel<<<(NN + 255) / 256, 256, 0, stream>>>(dinv, NN);

    // ---- layer 1: h = X @ W1 ; agg -> buf1 ----
    gemm_wmma_f32_kernel<0><<<gemm_blocks(NN, 256), 32 * WPB, 0, stream>>>(X, W1, buf0, NN, 256);
    self_bias_kernel<256><<<(NN * 256 + 255) / 256, 256, 0, stream>>>(buf0, dinv, b1, buf1, NN);
    {
        constexpr int EPB = 256 / (256 / 4);
        scatter_kernel<256><<<(E + EPB - 1) / EPB, 256, 0, stream>>>(esrc, edst, dinv, buf0, buf1, E);
    }
    // ---- layer 2: h = relu(buf1) @ W2 ; agg -> buf1 ----
    gemm_wmma_f32_kernel<1><<<gemm_blocks(NN, 256), 32 * WPB, 0, stream>>>(buf1, W2, buf0, NN, 256);
    self_bias_kernel<256><<<(NN * 256 + 255) / 256, 256, 0, stream>>>(buf0, dinv, b2, buf1, NN);
    {
        constexpr int EPB = 256 / (256 / 4);
        scatter_kernel<256><<<(E + EPB - 1) / EPB, 256, 0, stream>>>(esrc, edst, dinv, buf0, buf1, E);
    }
    // ---- layer 3: h = relu(buf1) @ W3 ; agg -> out (no relu) ----
    gemm_wmma_f32_kernel<1><<<gemm_blocks(NN, 128), 32 * WPB, 0, stream>>>(buf1, W3, buf0, NN, 128);
    self_bias_kernel<128><<<(NN * 128 + 255) / 256, 256, 0, stream>>>(buf0, dinv, b3, out, NN);
    {
        constexpr int EPB = 256 / (128 / 4);
        scatter_kernel<128><<<(E + EPB - 1) / EPB, 256, 0, stream>>>(esrc, edst, dinv, buf0, out, E);
    }
}

extern "C" void kernel_launch(void* const* d_in, const int* in_sizes, int n_in,
                              void* d_out, int out_size, void* d_ws, size_t ws_size,
                              hipStream_t stream) {
    const int E = in_sizes[1] / 2;

    // workspace layout: buf0 (NN*256), buf1 (NN*256), dinv (NN) — reused across graphs
    float* buf0 = (float*)d_ws;
    float* buf1 = buf0 + (size_t)NN * 256;
    float* dinv = buf1 + (size_t)NN * 256;

    for (int g = 0; g < 2; ++g) {
        const float* X  = (const float*)d_in[g == 0 ? 0 : 2];
        const int*   ei = (const int*)  d_in[g == 0 ? 1 : 3];
        const int wb = 4 + 6 * g;
        const float* W1 = (const float*)d_in[wb + 0];
        const float* b1 = (const float*)d_in[wb + 1];
        const float* W2 = (const float*)d_in[wb + 2];
        const float* b2 = (const float*)d_in[wb + 3];
        const float* W3 = (const float*)d_in[wb + 4];
        const float* b3 = (const float*)d_in[wb + 5];
        float* outg = (float*)d_out + (size_t)g * NN * 128;

        gcn_graph(X, ei, E, W1, b1, W2, b2, W3, b3, outg, buf0, buf1, dinv, stream);
    }
}